// DiffGraphAttentionLayer_13331578487355
// MI455X (gfx1250) — compile-verified
//
#include <hip/hip_runtime.h>
#include <hip/hip_bf16.h>

#define B_     4
#define N_     2048
#define IN_F_  256
#define OUT_F_ 128
#define D_     64
#define NUM_REL_ 6
#define REL_DIM_ 10
#define LAMBDA_INIT_ 0.35550906759096926f   // 0.8 - 0.6*exp(-0.3)

typedef float v2f __attribute__((ext_vector_type(2)));
typedef float v8f __attribute__((ext_vector_type(8)));

// ---------------------------------------------------------------------------
// Wave32 reduction helpers
// ---------------------------------------------------------------------------
__device__ inline float waveMax(float v) {
    #pragma unroll
    for (int o = 16; o > 0; o >>= 1) v = fmaxf(v, __shfl_xor(v, o, 32));
    return v;
}
__device__ inline float waveSum(float v) {
    #pragma unroll
    for (int o = 16; o > 0; o >>= 1) v += __shfl_xor(v, o, 32);
    return v;
}

// ---------------------------------------------------------------------------
// Generic fp32 WMMA GEMM:  C[M x 128] = A[M x K] (row major) @ Bm[K x 128]
// Block = 256 threads = 8 waves (4 along M, 2 along O). Each wave owns a
// 16x64 tile -> 4x v8f accumulators, K-loop in steps of 4 via
// V_WMMA_F32_16X16X4_F32.
//
// Fragment layout assumptions (CDNA5 ISA 7.12.2):
//   A (16x4 f32):  lanes 0-15 rows M=0..15 holding K={k,k+1} in v0,v1;
//                  lanes 16-31 same rows holding K={k+2,k+3}.
//   B (4x16 f32):  mirrored: lanes 0-15 cols N=0..15 with K={k,k+1};
//                  lanes 16-31 K={k+2,k+3}.
//   C/D (16x16):   VGPR v: lanes 0-15 -> M=v, lanes 16-31 -> M=v+8.
// ---------------------------------------------------------------------------
__global__ __launch_bounds__(256) void gemm_f32_wmma(
    const float* __restrict__ A, const float* __restrict__ Bm,
    float* __restrict__ C,
    int lda, int K,
    long long sA, long long sB, long long sC, int tilesPerBatch)
{
    const int bx    = blockIdx.x;
    const int batch = bx / tilesPerBatch;
    const int tile  = bx % tilesPerBatch;
    A  += (long long)batch * sA;
    Bm += (long long)batch * sB;
    C  += (long long)batch * sC;

    const int tid   = threadIdx.x;
    const int lane  = tid & 31;
    const int wave  = tid >> 5;
    const int waveM = wave >> 1;          // 0..3
    const int waveO = wave & 1;           // 0..1
    const int row0  = tile * 64 + waveM * 16;
    const int col0  = waveO * 64;
    const int half  = lane >> 4;          // 0 or 1 (K sub-pair)
    const int l16   = lane & 15;

    const float* Ap = A  + (size_t)(row0 + l16) * lda + 2 * half;
    const float* Bp = Bm + (size_t)(2 * half) * OUT_F_ + col0 + l16;

    v8f acc0 = {}, acc1 = {}, acc2 = {}, acc3 = {};

    for (int k = 0; k < K; k += 4) {
        if ((k & 127) == 0 && k + 128 < K)
            __builtin_prefetch(Ap + k + 128, 0, 1);   // global_prefetch_b8

        v2f a = *(const v2f*)(Ap + k);                // K={k,k+1} or {k+2,k+3}

        const float* bp = Bp + (size_t)k * OUT_F_;
        v2f b0, b1, b2, b3;
        b0.x = bp[0];           b0.y = bp[OUT_F_];
        b1.x = bp[16];          b1.y = bp[16 + OUT_F_];
        b2.x = bp[32];          b2.y = bp[32 + OUT_F_];
        b3.x = bp[48];          b3.y = bp[48 + OUT_F_];

        acc0 = __builtin_amdgcn_wmma_f32_16x16x4_f32(false, a, false, b0,
                                                     (short)0, acc0, false, false);
        acc1 = __builtin_amdgcn_wmma_f32_16x16x4_f32(false, a, false, b1,
                                                     (short)0, acc1, false, false);
        acc2 = __builtin_amdgcn_wmma_f32_16x16x4_f32(false, a, false, b2,
                                                     (short)0, acc2, false, false);
        acc3 = __builtin_amdgcn_wmma_f32_16x16x4_f32(false, a, false, b3,
                                                     (short)0, acc3, false, false);
    }

    // Store C/D: row = row0 + v + 8*half, col = col0 + {0,16,32,48} + l16
    float* Cp = C + (size_t)(row0 + 8 * half) * OUT_F_ + col0 + l16;
    #pragma unroll
    for (int v = 0; v < 8; ++v) {
        Cp[(size_t)v * OUT_F_ +  0] = acc0[v];
        Cp[(size_t)v * OUT_F_ + 16] = acc1[v];
        Cp[(size_t)v * OUT_F_ + 32] = acc2[v];
        Cp[(size_t)v * OUT_F_ + 48] = acc3[v];
    }
}

// ---------------------------------------------------------------------------
// Per-row rank-1 scores + relation scores + lambda scalar
// scal layout: [0..5]=rel_score_pos, [6..11]=rel_score_neg, [12]=lambda_full
// ---------------------------------------------------------------------------
__global__ __launch_bounds__(256) void prep_kernel(
    const float* __restrict__ Wh,
    const float* __restrict__ alp, const float* __restrict__ arp,
    const float* __restrict__ aln, const float* __restrict__ arn,
    const float* __restrict__ rel_emb,
    const float* __restrict__ a_rel_pos, const float* __restrict__ a_rel_neg,
    const float* __restrict__ ll1, const float* __restrict__ lr1,
    const float* __restrict__ ll2, const float* __restrict__ lr2,
    float* __restrict__ el_pos, float* __restrict__ er_pos,
    float* __restrict__ el_neg, float* __restrict__ er_neg,
    float* __restrict__ scal)
{
    const int gid = blockIdx.x * 256 + threadIdx.x;   // 0 .. B*N-1
    const float* w = Wh + (size_t)gid * OUT_F_;
    float slp = 0.f, srp = 0.f, sln = 0.f, srn = 0.f;
    #pragma unroll 8
    for (int d = 0; d < D_; ++d) {
        float p = w[d], n = w[D_ + d];
        slp += p * alp[d]; srp += p * arp[d];
        sln += n * aln[d]; srn += n * arn[d];
    }
    el_pos[gid] = slp; er_pos[gid] = srp;
    el_neg[gid] = sln; er_neg[gid] = srn;

    if (gid == 0) {
        for (int r = 0; r < NUM_REL_; ++r) {
            float sp = 0.f, sn = 0.f;
            for (int d = 0; d < REL_DIM_; ++d) {
                sp += rel_emb[r * REL_DIM_ + d] * a_rel_pos[d];
                sn += rel_emb[r * REL_DIM_ + d] * a_rel_neg[d];
            }
            scal[r] = sp; scal[NUM_REL_ + r] = sn;
        }
        float s1 = 0.f, s2 = 0.f;
        for (int d = 0; d < D_; ++d) { s1 += ll1[d] * lr1[d]; s2 += ll2[d] * lr2[d]; }
        scal[2 * NUM_REL_] = __expf(s1) - __expf(s2) + LAMBDA_INIT_;
    }
}

// ---------------------------------------------------------------------------
// One block per (b,i) row: logits -> LDS, dual masked softmax, write
// attention = attn_pos - lambda * attn_neg  straight to output.
// ---------------------------------------------------------------------------
__global__ __launch_bounds__(256) void attn_kernel(
    const int*   __restrict__ adj,
    const float* __restrict__ el_pos, const float* __restrict__ er_pos,
    const float* __restrict__ el_neg, const float* __restrict__ er_neg,
    const float* __restrict__ scal,
    float* __restrict__ attn_out)
{
    __shared__ float sp[N_];
    __shared__ float sn[N_];
    __shared__ float redA[8], redB[8];
    __shared__ float bc[4];
    __shared__ float s_rsp[NUM_REL_], s_rsn[NUM_REL_];
    __shared__ float s_lam;

    const int tid  = threadIdx.x;
    const int row  = blockIdx.x;          // b*N + i
    const int b    = row >> 11;           // / 2048
    const int lane = tid & 31;
    const int wid  = tid >> 5;

    if (tid < NUM_REL_) { s_rsp[tid] = scal[tid]; s_rsn[tid] = scal[NUM_REL_ + tid]; }
    if (tid == NUM_REL_) s_lam = scal[2 * NUM_REL_];

    const int*   arow = adj    + (size_t)row * N_;
    const float* erp  = er_pos + (size_t)b * N_;
    const float* ern  = er_neg + (size_t)b * N_;
    const float  elp  = el_pos[row];
    const float  eln  = el_neg[row];
    __syncthreads();

    // Pass 1: masked leaky-relu logits into LDS, track maxima
    float lmp = -3.4e38f, lmn = -3.4e38f;
    for (int j = tid; j < N_; j += 256) {
        __builtin_prefetch(arow + j + 1024, 0, 1);
        int   r  = arow[j];
        float ep = elp + erp[j] + s_rsp[r];
        float en = eln + ern[j] + s_rsn[r];
        ep = (ep >= 0.f) ? ep : 0.2f * ep;
        en = (en >= 0.f) ? en : 0.2f * en;
        bool m = (r > 0);
        ep = m ? ep : -9.0e15f;
        en = m ? en : -9.0e15f;
        sp[j] = ep; sn[j] = en;
        lmp = fmaxf(lmp, ep); lmn = fmaxf(lmn, en);
    }
    lmp = waveMax(lmp); lmn = waveMax(lmn);
    if (lane == 0) { redA[wid] = lmp; redB[wid] = lmn; }
    __syncthreads();
    if (tid == 0) {
        float mp = redA[0], mn = redB[0];
        #pragma unroll
        for (int k = 1; k < 8; ++k) { mp = fmaxf(mp, redA[k]); mn = fmaxf(mn, redB[k]); }
        bc[0] = mp; bc[1] = mn;
    }
    __syncthreads();
    const float mp = bc[0], mn = bc[1];

    // Pass 2: exponentials + sums
    float lsp = 0.f, lsn = 0.f;
    for (int j = tid; j < N_; j += 256) {
        float ep = __expf(sp[j] - mp);
        float en = __expf(sn[j] - mn);
        sp[j] = ep; sn[j] = en;
        lsp += ep; lsn += en;
    }
    lsp = waveSum(lsp); lsn = waveSum(lsn);
    if (lane == 0) { redA[wid] = lsp; redB[wid] = lsn; }
    __syncthreads();
    if (tid == 0) {
        float a = 0.f, c = 0.f;
        #pragma unroll
        for (int k = 0; k < 8; ++k) { a += redA[k]; c += redB[k]; }
        bc[2] = 1.f / a; bc[3] = 1.f / c;
    }
    __syncthreads();
    const float isp = bc[2], isn = bc[3], lam = s_lam;

    // Pass 3: write combined attention row
    float* orow = attn_out + (size_t)row * N_;
    for (int j = tid; j < N_; j += 256)
        orow[j] = sp[j] * isp - lam * (sn[j] * isn);
}

// ---------------------------------------------------------------------------
// In-place LayerNorm + (1-LAMBDA_INIT) scale + exact GELU, 128-wide rows
// ---------------------------------------------------------------------------
__global__ __launch_bounds__(128) void ln_gelu_kernel(
    float* __restrict__ io,
    const float* __restrict__ gamma, const float* __restrict__ beta)
{
    __shared__ float red[4];
    __shared__ float bcv;
    const int tid  = threadIdx.x;
    const int row  = blockIdx.x;
    const int lane = tid & 31;
    const int wid  = tid >> 5;

    float x = io[(size_t)row * OUT_F_ + tid];

    float s = waveSum(x);
    if (lane == 0) red[wid] = s;
    __syncthreads();
    if (tid == 0) bcv = (red[0] + red[1] + red[2] + red[3]) * (1.f / OUT_F_);
    __syncthreads();
    const float mu = bcv;
    const float d  = x - mu;

    float s2 = waveSum(d * d);
    __syncthreads();
    if (lane == 0) red[wid] = s2;
    __syncthreads();
    if (tid == 0) bcv = (red[0] + red[1] + red[2] + red[3]) * (1.f / OUT_F_);
    __syncthreads();
    const float var = bcv;

    float y = d * rsqrtf(var + 1e-5f) * gamma[tid] + beta[tid];
    y *= (1.0f - LAMBDA_INIT_);
    float g = 0.5f * y * (1.0f + erff(y * 0.70710678118654752f));
    io[(size_t)row * OUT_F_ + tid] = g;
}

// ---------------------------------------------------------------------------
extern "C" void kernel_launch(void* const* d_in, const int* in_sizes, int n_in,
                              void* d_out, int out_size, void* d_ws, size_t ws_size,
                              hipStream_t stream)
{
    const float* h        = (const float*)d_in[0];
    const int*   adj      = (const int*)  d_in[1];
    const float* W        = (const float*)d_in[2];
    const float* alp      = (const float*)d_in[3];
    const float* arp      = (const float*)d_in[4];
    const float* aln      = (const float*)d_in[5];
    const float* arn      = (const float*)d_in[6];
    const float* rel_emb  = (const float*)d_in[7];
    const float* a_rel_p  = (const float*)d_in[8];
    const float* a_rel_n  = (const float*)d_in[9];
    const float* ll1      = (const float*)d_in[10];
    const float* lr1      = (const float*)d_in[11];
    const float* ll2      = (const float*)d_in[12];
    const float* lr2      = (const float*)d_in[13];
    const float* ln_gamma = (const float*)d_in[14];
    const float* ln_beta  = (const float*)d_in[15];

    float* out      = (float*)d_out;
    float* out_h    = out;                                    // B*N*128
    float* out_attn = out + (size_t)B_ * N_ * OUT_F_;          // B*N*N

    float* ws      = (float*)d_ws;
    float* Wh      = ws;                                       // B*N*128
    float* el_pos  = ws + (size_t)B_ * N_ * OUT_F_;
    float* er_pos  = el_pos + B_ * N_;
    float* el_neg  = er_pos + B_ * N_;
    float* er_neg  = el_neg + B_ * N_;
    float* scal    = er_neg + B_ * N_;                         // 13 floats

    // 1) Wh = h @ W          (8192 x 256) @ (256 x 128)
    gemm_f32_wmma<<<(B_ * N_) / 64, 256, 0, stream>>>(
        h, W, Wh, IN_F_, IN_F_, 0LL, 0LL, 0LL, (B_ * N_) / 64);

    // 2) per-row scores + rel scores + lambda
    prep_kernel<<<(B_ * N_) / 256, 256, 0, stream>>>(
        Wh, alp, arp, aln, arn, rel_emb, a_rel_p, a_rel_n,
        ll1, lr1, ll2, lr2, el_pos, er_pos, el_neg, er_neg, scal);

    // 3) attention rows (written directly into the output tensor)
    attn_kernel<<<B_ * N_, 256, 0, stream>>>(
        adj, el_pos, er_pos, el_neg, er_neg, scal, out_attn);

    // 4) h_prime = attention @ Wh   per batch: (2048x2048) @ (2048x128)
    gemm_f32_wmma<<<B_ * (N_ / 64), 256, 0, stream>>>(
        out_attn, Wh, out_h, N_, N_,
        (long long)N_ * N_, (long long)N_ * OUT_F_, (long long)N_ * OUT_F_,
        N_ / 64);

    // 5) LayerNorm + GELU in place on out_h
    ln_gelu_kernel<<<B_ * N_, 128, 0, stream>>>(out_h, ln_gamma, ln_beta);
}